// PyGTemporalGNN_19241453486692
// MI455X (gfx1250) — compile-verified
//
#include <hip/hip_runtime.h>

#define N_NODES  100000
#define N_EDGES  1600000
#define BATCH    8192
#define D        128
#define N_TXF    16
#define D_TX     32
#define D_CLS_IN 288
#define D_CLS_H  64

#define NTILES   (N_NODES / 16)     // 6250
#define SAGE_BLOCKS 1024

// padded LDS row strides (bank-conflict-free for 16-lane b64 fragment reads)
#define DP       (D + 4)            // 132: lane->bank = 4*l15
#define ZP       (D_CLS_IN + 4)     // 292: lane*36 mod 64 all-distinct
#define HP       (D_CLS_H + 4)      // 68:  lane->bank = 4*l15

typedef __attribute__((ext_vector_type(2))) float v2f;
typedef __attribute__((ext_vector_type(8))) float v8f;

#if defined(__gfx1250__) && __has_builtin(__builtin_amdgcn_global_load_async_to_lds_b128)
#define HAVE_ASYNC_LDS 1
typedef int v4i_t __attribute__((vector_size(16)));
typedef __attribute__((address_space(1))) v4i_t as1_v4i;
typedef __attribute__((address_space(3))) v4i_t as3_v4i;
#else
#define HAVE_ASYNC_LDS 0
#endif

// ---------------------------------------------------------------- utilities
__global__ void zero_f32(float* __restrict__ p, int n) {
    int i = blockIdx.x * blockDim.x + threadIdx.x;
    if (i < n) p[i] = 0.0f;
}

// h0[i][:] = emb_table[ids[i]][:]   (one float4 per thread, 32 threads/row)
__global__ void gather_embed(float* __restrict__ h,
                             const float* __restrict__ emb,
                             const int* __restrict__ ids) {
    int t = blockIdx.x * blockDim.x + threadIdx.x;
    int node = t >> 5;
    if (node >= N_NODES) return;
    int c4 = (t & 31) << 2;
    float4 v = *(const float4*)(emb + (size_t)ids[node] * D + c4);
    *(float4*)(h + (size_t)node * D + c4) = v;
}

// cnt[dst] += 1
__global__ void degree_count(float* __restrict__ cnt, const int* __restrict__ ei) {
    int e = blockIdx.x * blockDim.x + threadIdx.x;
    if (e < N_EDGES) atomicAdd(cnt + ei[N_EDGES + e], 1.0f);
}

// msg[dst][:] += h[src][:]   (one wave per edge, float4 per lane, L2 atomics)
__global__ void scatter_add(float* __restrict__ msg,
                            const float* __restrict__ h,
                            const int* __restrict__ ei) {
    size_t t = (size_t)blockIdx.x * blockDim.x + threadIdx.x;
    size_t e = t >> 5;
    if (e >= N_EDGES) return;
    int c4 = ((int)t & 31) << 2;
    int src = ei[e];
    int dst = ei[N_EDGES + e];
    float4 v = *(const float4*)(h + (size_t)src * D + c4);
    float* o = msg + (size_t)dst * D + c4;
    atomicAdd(o + 0, v.x);
    atomicAdd(o + 1, v.y);
    atomicAdd(o + 2, v.z);
    atomicAdd(o + 3, v.w);
}

// out = relu( (msg@Wl)*inv + bl + h@Wr )  (mean folded into epilogue)
// persistent weight-stationary: each wave keeps its 16-col slice of Wl/Wr in
// registers (64 v2f fragments) and grid-strides over 16-row tiles.
// Tiles staged straight into LDS with async DMA when available.
__global__ __launch_bounds__(256)
void sage_wmma(float* __restrict__ out,
               const float* __restrict__ msg,
               const float* __restrict__ cnt,
               const float* __restrict__ h,
               const float* __restrict__ Wl,
               const float* __restrict__ bl,
               const float* __restrict__ Wr) {
    __shared__ float sA[16 * DP];   // raw neighbor-sum tile (padded)
    __shared__ float sH[16 * DP];   // self-feature tile (padded)

    int lane  = threadIdx.x & 31;
    int wave  = threadIdx.x >> 5;      // 0..7 -> 16-col tile
    int n0    = wave * 16;
    int l15   = lane & 15;             // A row / B,C column
    int khalf = (lane >> 4) << 1;      // K offset: 0 (lanes 0-15) or 2 (16-31)
    int mbase = (lane >> 4) << 3;      // C rows: 0 or 8

    // ---- load weight fragments once (stationary across all row tiles)
    v2f fl[32], fr[32];
#pragma unroll
    for (int kk = 0; kk < 32; ++kk) {
        int k0 = kk * 4 + khalf;
        fl[kk].x = Wl[(size_t)k0 * D + n0 + l15];
        fl[kk].y = Wl[(size_t)(k0 + 1) * D + n0 + l15];
        fr[kk].x = Wr[(size_t)k0 * D + n0 + l15];
        fr[kk].y = Wr[(size_t)(k0 + 1) * D + n0 + l15];
    }
    float bias = bl[n0 + l15];

    for (int tile = blockIdx.x; tile < NTILES; tile += gridDim.x) {
        int m0 = tile * 16;

        // prefetch next tile's A data while this tile computes
        int nt = tile + gridDim.x;
        if (nt < NTILES && threadIdx.x < 64) {
            int r = threadIdx.x & 15;
            const float* base = (threadIdx.x & 16) ? h : msg;
            int half = ((threadIdx.x >> 5) & 1) << 6;   // 0 or 64 floats
            __builtin_prefetch(base + (size_t)(nt * 16 + r) * D + half, 0, 0);
        }

        // ---- stage 16x128 raw msg + self tiles
#if HAVE_ASYNC_LDS
        for (int i = threadIdx.x; i < 16 * 32; i += 256) {
            int r  = i >> 5;
            int c4 = (i & 31) << 2;
            const float* gm = msg + (size_t)(m0 + r) * D + c4;
            const float* gh = h + (size_t)(m0 + r) * D + c4;
            __builtin_amdgcn_global_load_async_to_lds_b128(
                (as1_v4i*)gm, (as3_v4i*)(sA + r * DP + c4), 0, 0);
            __builtin_amdgcn_global_load_async_to_lds_b128(
                (as1_v4i*)gh, (as3_v4i*)(sH + r * DP + c4), 0, 0);
        }
        asm volatile("s_wait_asynccnt 0x0" ::: "memory");
#else
        for (int i = threadIdx.x; i < 16 * 32; i += 256) {
            int r  = i >> 5;
            int c4 = (i & 31) << 2;
            float4 vm = *(const float4*)(msg + (size_t)(m0 + r) * D + c4);
            float4 vh = *(const float4*)(h + (size_t)(m0 + r) * D + c4);
            *(float4*)(sA + r * DP + c4) = vm;
            *(float4*)(sH + r * DP + c4) = vh;
        }
#endif
        __syncthreads();

        // ---- 64 WMMAs, weights from registers, A from LDS
        v8f accL = {}, accR = {};
#pragma unroll
        for (int kk = 0; kk < 32; ++kk) {
            v2f a;
            a.x = sA[l15 * DP + kk * 4 + khalf];
            a.y = sA[l15 * DP + kk * 4 + khalf + 1];
            accL = __builtin_amdgcn_wmma_f32_16x16x4_f32(false, a, false, fl[kk],
                                                         (short)0, accL, false, false);
        }
#pragma unroll
        for (int kk = 0; kk < 32; ++kk) {
            v2f a;
            a.x = sH[l15 * DP + kk * 4 + khalf];
            a.y = sH[l15 * DP + kk * 4 + khalf + 1];
            accR = __builtin_amdgcn_wmma_f32_16x16x4_f32(false, a, false, fr[kk],
                                                         (short)0, accR, false, false);
        }

        // ---- epilogue: mean scale, +bias, relu, store
#pragma unroll
        for (int r = 0; r < 8; ++r) {
            int row = m0 + mbase + r;
            float inv = 1.0f / fmaxf(cnt[row], 1.0f);
            float v = fmaxf(accL[r] * inv + accR[r] + bias, 0.0f);
            out[(size_t)row * D + n0 + l15] = v;
        }
        __syncthreads();
    }
}

// classifier: z = [h2[s] | h2[r] | relu(txf@Wt+bt)];  relu(z@Wc1+bc1) @ Wc2 + bc2
// block = 128 threads = 4 waves; handles 16 batch rows
__global__ __launch_bounds__(128)
void classifier_wmma(float* __restrict__ out,
                     const float* __restrict__ h2,
                     const int* __restrict__ sidx,
                     const int* __restrict__ ridx,
                     const float* __restrict__ txf,
                     const float* __restrict__ Wt,
                     const float* __restrict__ bt,
                     const float* __restrict__ Wc1,
                     const float* __restrict__ bc1,
                     const float* __restrict__ Wc2,
                     const float* __restrict__ bc2) {
    __shared__ float sZ[16 * ZP];    // concat input rows (padded)
    __shared__ float sZH[16 * HP];   // hidden activations (padded)
    int r0 = blockIdx.x * 16;

    // build the 16 x 288 concat block (tx projection computed inline)
    for (int i = threadIdx.x; i < 16 * D_CLS_IN; i += 128) {
        int r = i / D_CLS_IN;
        int c = i - r * D_CLS_IN;
        int row = r0 + r;
        float v;
        if (c < D) {
            v = h2[(size_t)sidx[row] * D + c];
        } else if (c < 2 * D) {
            v = h2[(size_t)ridx[row] * D + (c - D)];
        } else {
            int j = c - 2 * D;
            float s = bt[j];
#pragma unroll
            for (int k = 0; k < N_TXF; ++k)
                s += txf[row * N_TXF + k] * Wt[k * D_TX + j];
            v = fmaxf(s, 0.0f);
        }
        sZ[r * ZP + c] = v;
    }
    __syncthreads();

    int lane  = threadIdx.x & 31;
    int wave  = threadIdx.x >> 5;      // 0..3 -> 16-col tile of the 64 hidden
    int n0    = wave * 16;
    int l15   = lane & 15;
    int khalf = (lane >> 4) << 1;

    v8f acc = {};
    for (int kb = 0; kb < D_CLS_IN; kb += 4) {
        int k0 = kb + khalf;
        v2f a, b;
        a.x = sZ[l15 * ZP + k0];
        a.y = sZ[l15 * ZP + k0 + 1];
        b.x = Wc1[(size_t)k0 * D_CLS_H + n0 + l15];
        b.y = Wc1[(size_t)(k0 + 1) * D_CLS_H + n0 + l15];
        acc = __builtin_amdgcn_wmma_f32_16x16x4_f32(false, a, false, b,
                                                    (short)0, acc, false, false);
    }
    float bias = bc1[n0 + l15];
    int mbase = (lane >> 4) << 3;
#pragma unroll
    for (int r = 0; r < 8; ++r)
        sZH[(mbase + r) * HP + n0 + l15] = fmaxf(acc[r] + bias, 0.0f);
    __syncthreads();

    // final 64 -> 1 projection
    if (threadIdx.x < 16) {
        float s = bc2[0];
#pragma unroll 8
        for (int j = 0; j < D_CLS_H; ++j)
            s += sZH[threadIdx.x * HP + j] * Wc2[j];
        out[r0 + threadIdx.x] = s;
    }
}

// ---------------------------------------------------------------- launcher
extern "C" void kernel_launch(void* const* d_in, const int* in_sizes, int n_in,
                              void* d_out, int out_size, void* d_ws, size_t ws_size,
                              hipStream_t stream) {
    const int*   ids  = (const int*)d_in[0];
    const int*   ei   = (const int*)d_in[1];
    const int*   sidx = (const int*)d_in[2];
    const int*   ridx = (const int*)d_in[3];
    const float* txf  = (const float*)d_in[4];
    const float* emb  = (const float*)d_in[5];
    const float* Wl1  = (const float*)d_in[6];
    const float* bl1  = (const float*)d_in[7];
    const float* Wr1  = (const float*)d_in[8];
    const float* Wl2  = (const float*)d_in[9];
    const float* bl2  = (const float*)d_in[10];
    const float* Wr2  = (const float*)d_in[11];
    const float* Wt   = (const float*)d_in[12];
    const float* bt   = (const float*)d_in[13];
    const float* Wc1  = (const float*)d_in[14];
    const float* bc1  = (const float*)d_in[15];
    const float* Wc2  = (const float*)d_in[16];
    const float* bc2  = (const float*)d_in[17];
    float* outp = (float*)d_out;

    const size_t SZ = (size_t)N_NODES * D;     // 12.8M floats = 51.2 MB
    float* buf0 = (float*)d_ws;                // h0, later h2
    float* buf1 = buf0 + SZ;                   // msg accumulator
    float* buf2 = buf1 + SZ;                   // h1
    float* cnt  = buf2 + SZ;                   // degrees

    const int ZB = 256;
    int scatterBlocks = (int)(((size_t)N_EDGES * 32 + ZB - 1) / ZB);

    // init (re-done every call: graph-replay safe)
    zero_f32<<<(N_NODES + ZB - 1) / ZB, ZB, 0, stream>>>(cnt, N_NODES);
    zero_f32<<<(int)((SZ + ZB - 1) / ZB), ZB, 0, stream>>>(buf1, (int)SZ);

    gather_embed<<<(N_NODES * 32 + ZB - 1) / ZB, ZB, 0, stream>>>(buf0, emb, ids);
    degree_count<<<(N_EDGES + ZB - 1) / ZB, ZB, 0, stream>>>(cnt, ei);

    // layer 1
    scatter_add<<<scatterBlocks, ZB, 0, stream>>>(buf1, buf0, ei);
    sage_wmma<<<SAGE_BLOCKS, 256, 0, stream>>>(buf2, buf1, cnt, buf0, Wl1, bl1, Wr1);

    // layer 2
    zero_f32<<<(int)((SZ + ZB - 1) / ZB), ZB, 0, stream>>>(buf1, (int)SZ);
    scatter_add<<<scatterBlocks, ZB, 0, stream>>>(buf1, buf2, ei);
    sage_wmma<<<SAGE_BLOCKS, 256, 0, stream>>>(buf0, buf1, cnt, buf2, Wl2, bl2, Wr2);

    // classifier head
    classifier_wmma<<<BATCH / 16, 128, 0, stream>>>(outp, buf0, sidx, ridx, txf,
                                                    Wt, bt, Wc1, bc1, Wc2, bc2);
}